// ReactionGAT_20005957664861
// MI455X (gfx1250) — compile-verified
//
#include <hip/hip_runtime.h>
#include <math.h>

#define NNODES 50000
#define NEDGES 800000
#define EPSV   1e-16f

typedef float v2f __attribute__((ext_vector_type(2)));
typedef float v8f __attribute__((ext_vector_type(8)));

__device__ __forceinline__ float lrelu02(float x) { return x > 0.f ? x : 0.2f * x; }

// float atomic max via signed/unsigned int ordering trick
__device__ __forceinline__ void atomicMaxF(float* addr, float val) {
    if (val >= 0.f) atomicMax((int*)addr, __float_as_int(val));
    else            atomicMin((unsigned int*)addr, __float_as_uint(val));
}

__device__ __forceinline__ float waveReduceSum(float v) {
#pragma unroll
    for (int off = 16; off > 0; off >>= 1) v += __shfl_xor(v, off, 32);
    return v;
}

// ---------------------------------------------------------------------------
// FP32 WMMA GEMM: C[M,Nn] = A[M,K] * B[K,Nn] (+ optional per-column bias)
// One wave computes one 16x16 tile via K/4 chained V_WMMA_F32_16X16X4_F32.
// A layout (16x4 MxK): lanes 0-15 hold K={k0,k0+1}, lanes 16-31 K={k0+2,k0+3}
// C/D layout: VGPR v -> row v (lanes 0-15) / row v+8 (lanes 16-31), col = lane&15
// ---------------------------------------------------------------------------
template <int K>
__global__ void gemm_wmma_f32(const float* __restrict__ A, const float* __restrict__ B,
                              const float* __restrict__ bias, float* __restrict__ C,
                              int M, int Nn) {
    const int wavesPerBlock = blockDim.x >> 5;
    const int wid  = blockIdx.x * wavesPerBlock + (threadIdx.x >> 5);
    const int ntn  = Nn >> 4;
    const int tiles = (M >> 4) * ntn;
    if (wid >= tiles) return;                    // wave-uniform exit, EXEC stays full
    const int tm   = wid / ntn;
    const int tn   = wid % ntn;
    const int lane = threadIdx.x & 31;
    const int half = lane >> 4;
    const int l15  = lane & 15;

    v8f acc = {0.f, 0.f, 0.f, 0.f, 0.f, 0.f, 0.f, 0.f};
    const float* Ar = A + (size_t)(tm * 16 + l15) * K;
    const float* Bc = B + tn * 16 + l15;

#pragma unroll 4
    for (int k0 = 0; k0 < K; k0 += 4) {
        const int ka = k0 + 2 * half;
        v2f a, b;
        a.x = Ar[ka];
        a.y = Ar[ka + 1];
        b.x = Bc[(size_t)ka * Nn];
        b.y = Bc[(size_t)(ka + 1) * Nn];
        acc = __builtin_amdgcn_wmma_f32_16x16x4_f32(false, a, false, b,
                                                    (short)0, acc, false, false);
    }

    const float badd = bias ? bias[tn * 16 + l15] : 0.f;
    float* Cp = C + (size_t)(tm * 16 + half * 8) * Nn + tn * 16 + l15;
#pragma unroll
    for (int v = 0; v < 8; ++v) Cp[(size_t)v * Nn] = acc[v] + badd;
}

// ---------------------------------------------------------------------------
// al_src[n,hd] = sum_c h[n,hd,c]*a_src[hd,c]  (and same for dst). Wave per node.
// ---------------------------------------------------------------------------
template <int H, int C>
__global__ void gat_node_alphas(const float* __restrict__ h,
                                const float* __restrict__ a_src,
                                const float* __restrict__ a_dst,
                                float* __restrict__ als, float* __restrict__ ald, int n) {
    const int nd = blockIdx.x * (blockDim.x >> 5) + (threadIdx.x >> 5);
    if (nd >= n) return;
    const int lane = threadIdx.x & 31;
    const float* hr = h + (size_t)nd * (H * C);
#pragma unroll
    for (int hd = 0; hd < H; ++hd) {
        float ps = 0.f, pd = 0.f;
#pragma unroll
        for (int cc = lane; cc < C; cc += 32) {
            const float hv = hr[hd * C + cc];
            ps += hv * a_src[hd * C + cc];
            pd += hv * a_dst[hd * C + cc];
        }
        ps = waveReduceSum(ps);
        pd = waveReduceSum(pd);
        if (lane == 0) { als[nd * H + hd] = ps; ald[nd * H + hd] = pd; }
    }
}

// m init with the (always-present) self-loop logit
template <int H>
__global__ void gat_node_selfmax(const float* __restrict__ als, const float* __restrict__ ald,
                                 float* __restrict__ m, int n) {
    const int i = blockIdx.x * blockDim.x + threadIdx.x;
    if (i < n * H) m[i] = lrelu02(als[i] + ald[i]);
}

// running max over incoming edges
template <int H>
__global__ void gat_edge_max(const int* __restrict__ row, const int* __restrict__ col,
                             const float* __restrict__ als, const float* __restrict__ ald,
                             float* __restrict__ m, int E) {
    const int e = blockIdx.x * blockDim.x + threadIdx.x;
    if (e >= E) return;
    const int r = row[e], c = col[e];
#pragma unroll
    for (int hd = 0; hd < H; ++hd)
        atomicMaxF(&m[c * H + hd], lrelu02(als[r * H + hd] + ald[c * H + hd]));
}

// s init with self-loop exp term (uses finalized m)
template <int H>
__global__ void gat_node_selfexp(const float* __restrict__ als, const float* __restrict__ ald,
                                 const float* __restrict__ m, float* __restrict__ s, int n) {
    const int i = blockIdx.x * blockDim.x + threadIdx.x;
    if (i < n * H) s[i] = __expf(lrelu02(als[i] + ald[i]) - m[i]);
}

// per-edge weight w = exp(e - m[col]); accumulate denominator s[col]
template <int H>
__global__ void gat_edge_sumexp(const int* __restrict__ row, const int* __restrict__ col,
                                const float* __restrict__ als, const float* __restrict__ ald,
                                const float* __restrict__ m, float* __restrict__ w,
                                float* __restrict__ s, int E) {
    const int e = blockIdx.x * blockDim.x + threadIdx.x;
    if (e >= E) return;
    const int r = row[e], c = col[e];
#pragma unroll
    for (int hd = 0; hd < H; ++hd) {
        const float v  = lrelu02(als[r * H + hd] + ald[c * H + hd]);
        const float wv = __expf(v - m[c * H + hd]);
        w[(size_t)e * H + hd] = wv;
        atomicAdd(&s[c * H + hd], wv);
    }
}

// wave per edge: agg[col] += h[row] * alpha  (alpha = w/(s+eps), per head)
template <int H, int C>
__global__ void gat_edge_aggregate(const int* __restrict__ row, const int* __restrict__ col,
                                   const float* __restrict__ h, const float* __restrict__ w,
                                   const float* __restrict__ s, float* __restrict__ agg, int E) {
    const int e = blockIdx.x * (blockDim.x >> 5) + (threadIdx.x >> 5);
    if (e >= E) return;
    const int lane = threadIdx.x & 31;
    const int r = row[e], c = col[e];
    float alpha[H];
#pragma unroll
    for (int hd = 0; hd < H; ++hd)
        alpha[hd] = w[(size_t)e * H + hd] / (s[c * H + hd] + EPSV);
    const float* hr  = h   + (size_t)r * (H * C);
    float*       ag  = agg + (size_t)c * (H * C);
#pragma unroll
    for (int j = lane; j < H * C; j += 32)
        atomicAdd(&ag[j], hr[j] * alpha[j / C]);
}

// add self-loop message + bias, ELU; writes activated output over agg in place
template <int H, int C>
__global__ void gat_node_epilogue(const float* __restrict__ h,
                                  const float* __restrict__ als, const float* __restrict__ ald,
                                  const float* __restrict__ m, const float* __restrict__ s,
                                  const float* __restrict__ b, float* __restrict__ agg, int n) {
    const int idx = blockIdx.x * blockDim.x + threadIdx.x;
    if (idx >= n * H * C) return;
    const int nd = idx / (H * C);
    const int j  = idx % (H * C);
    const int hd = j / C;
    float es = lrelu02(als[nd * H + hd] + ald[nd * H + hd]);
    const float alpha_self = __expf(es - m[nd * H + hd]) / (s[nd * H + hd] + EPSV);
    float o = agg[idx] + h[idx] * alpha_self + b[j];
    agg[idx] = o > 0.f ? o : __expf(o) - 1.f;   // ELU
}

// per-node partial dots for the edge head: p_src = h·We[0:64], p_dst = h·We[64:128]
__global__ void edge_node_partials(const float* __restrict__ h, const float* __restrict__ We,
                                   float* __restrict__ ps, float* __restrict__ pd, int n) {
    const int nd = blockIdx.x * (blockDim.x >> 5) + (threadIdx.x >> 5);
    if (nd >= n) return;
    const int lane = threadIdx.x & 31;
    const float* hr = h + (size_t)nd * 64;
    float a = hr[lane] * We[lane]      + hr[lane + 32] * We[lane + 32];
    float b = hr[lane] * We[64 + lane] + hr[lane + 32] * We[96 + lane];
    a = waveReduceSum(a);
    b = waveReduceSum(b);
    if (lane == 0) { ps[nd] = a; pd[nd] = b; }
}

__global__ void edge_output_k(const int* __restrict__ row, const int* __restrict__ col,
                              const float* __restrict__ ea, const float* __restrict__ We,
                              const float* __restrict__ be, const float* __restrict__ ps,
                              const float* __restrict__ pd, float* __restrict__ out, int E) {
    const int e = blockIdx.x * blockDim.x + threadIdx.x;
    if (e >= E) return;
    float acc = ps[row[e]] + pd[col[e]] + be[0];
    const float4* a4 = (const float4*)(ea + (size_t)e * 16);
    const float4* w4 = (const float4*)(We + 128);
#pragma unroll
    for (int q = 0; q < 4; ++q) {
        const float4 av = a4[q], wv = w4[q];
        acc += av.x * wv.x + av.y * wv.y + av.z * wv.z + av.w * wv.w;
    }
    out[e] = acc;
}

// ---------------------------------------------------------------------------
extern "C" void kernel_launch(void* const* d_in, const int* in_sizes, int n_in,
                              void* d_out, int out_size, void* d_ws, size_t ws_size,
                              hipStream_t stream) {
    const float* x   = (const float*)d_in[0];
    const int*   ei  = (const int*)d_in[1];
    const int*   row = ei;                 // sources
    const int*   col = ei + NEDGES;        // destinations (softmax segments)
    const float* ea  = (const float*)d_in[2];
    const float* W1  = (const float*)d_in[3];
    const float* a1s = (const float*)d_in[4];
    const float* a1d = (const float*)d_in[5];
    const float* b1  = (const float*)d_in[6];
    const float* W2  = (const float*)d_in[7];
    const float* a2s = (const float*)d_in[8];
    const float* a2d = (const float*)d_in[9];
    const float* b2  = (const float*)d_in[10];
    const float* Wn  = (const float*)d_in[11];
    const float* bn  = (const float*)d_in[12];
    const float* We  = (const float*)d_in[13];
    const float* be  = (const float*)d_in[14];

    float* node_out = (float*)d_out;                      // [N,32]
    float* edge_out = node_out + (size_t)NNODES * 32;     // [E,1]

    // carve workspace (~148 MB)
    float* p    = (float*)d_ws;
    float* h1   = p; p += (size_t)NNODES * 256;   // layer-1 pre-agg features
    float* agg1 = p; p += (size_t)NNODES * 256;   // layer-1 aggregate -> h1act
    float* h2   = p; p += (size_t)NNODES * 64;    // layer-2 pre-agg features
    float* agg2 = p; p += (size_t)NNODES * 64;    // layer-2 aggregate -> h2act
    float* als1 = p; p += NNODES * 4;
    float* ald1 = p; p += NNODES * 4;
    float* m1   = p; p += NNODES * 4;
    float* s1   = p; p += NNODES * 4;
    float* als2 = p; p += NNODES;
    float* ald2 = p; p += NNODES;
    float* m2   = p; p += NNODES;
    float* s2   = p; p += NNODES;
    float* w1   = p; p += (size_t)NEDGES * 4;
    float* w2   = p; p += NEDGES;
    float* psrc = p; p += NNODES;
    float* pdst = p; p += NNODES;

    const int TB = 256;                 // 8 waves / block
    const int edgeBlocks  = (NEDGES + TB - 1) / TB;
    const int edgeWaveBlk = (NEDGES + 7) / 8;
    const int nodeWaveBlk = (NNODES + 7) / 8;

    // ---- Layer 1 (H=4, C=64) ----
    {
        const int tiles = (NNODES / 16) * (256 / 16);
        gemm_wmma_f32<64><<<(tiles + 7) / 8, TB, 0, stream>>>(x, W1, nullptr, h1, NNODES, 256);
    }
    hipMemsetAsync(agg1, 0, (size_t)NNODES * 256 * sizeof(float), stream);
    gat_node_alphas<4, 64><<<nodeWaveBlk, TB, 0, stream>>>(h1, a1s, a1d, als1, ald1, NNODES);
    gat_node_selfmax<4><<<(NNODES * 4 + TB - 1) / TB, TB, 0, stream>>>(als1, ald1, m1, NNODES);
    gat_edge_max<4><<<edgeBlocks, TB, 0, stream>>>(row, col, als1, ald1, m1, NEDGES);
    gat_node_selfexp<4><<<(NNODES * 4 + TB - 1) / TB, TB, 0, stream>>>(als1, ald1, m1, s1, NNODES);
    gat_edge_sumexp<4><<<edgeBlocks, TB, 0, stream>>>(row, col, als1, ald1, m1, w1, s1, NEDGES);
    gat_edge_aggregate<4, 64><<<edgeWaveBlk, TB, 0, stream>>>(row, col, h1, w1, s1, agg1, NEDGES);
    gat_node_epilogue<4, 64><<<((size_t)NNODES * 256 + TB - 1) / TB, TB, 0, stream>>>(
        h1, als1, ald1, m1, s1, b1, agg1, NNODES);                 // agg1 = ELU layer-1 out

    // ---- Layer 2 (H=1, C=64) ----
    {
        const int tiles = (NNODES / 16) * (64 / 16);
        gemm_wmma_f32<256><<<(tiles + 7) / 8, TB, 0, stream>>>(agg1, W2, nullptr, h2, NNODES, 64);
    }
    hipMemsetAsync(agg2, 0, (size_t)NNODES * 64 * sizeof(float), stream);
    gat_node_alphas<1, 64><<<nodeWaveBlk, TB, 0, stream>>>(h2, a2s, a2d, als2, ald2, NNODES);
    gat_node_selfmax<1><<<(NNODES + TB - 1) / TB, TB, 0, stream>>>(als2, ald2, m2, NNODES);
    gat_edge_max<1><<<edgeBlocks, TB, 0, stream>>>(row, col, als2, ald2, m2, NEDGES);
    gat_node_selfexp<1><<<(NNODES + TB - 1) / TB, TB, 0, stream>>>(als2, ald2, m2, s2, NNODES);
    gat_edge_sumexp<1><<<edgeBlocks, TB, 0, stream>>>(row, col, als2, ald2, m2, w2, s2, NEDGES);
    gat_edge_aggregate<1, 64><<<edgeWaveBlk, TB, 0, stream>>>(row, col, h2, w2, s2, agg2, NEDGES);
    gat_node_epilogue<1, 64><<<((size_t)NNODES * 64 + TB - 1) / TB, TB, 0, stream>>>(
        h2, als2, ald2, m2, s2, b2, agg2, NNODES);                 // agg2 = ELU layer-2 out

    // ---- Node head: node_out = h2act @ Wn + bn ----
    {
        const int tiles = (NNODES / 16) * (32 / 16);
        gemm_wmma_f32<64><<<(tiles + 7) / 8, TB, 0, stream>>>(agg2, Wn, bn, node_out, NNODES, 32);
    }

    // ---- Edge head via per-node partial dots ----
    edge_node_partials<<<nodeWaveBlk, TB, 0, stream>>>(agg2, We, psrc, pdst, NNODES);
    edge_output_k<<<edgeBlocks, TB, 0, stream>>>(row, col, ea, We, be, psrc, pdst,
                                                 edge_out, NEDGES);
}